// MS_SSIM_41480794145192
// MI455X (gfx1250) — compile-verified
//
#include <hip/hip_runtime.h>
#include <math.h>

typedef __attribute__((ext_vector_type(2))) float v2f;
typedef __attribute__((ext_vector_type(8))) float v8f;

#define WIN 11
#define SSIM_C1 1e-4f
#define SSIM_C2 9e-4f

__device__ __forceinline__ v8f wmma_f32_k4(v2f a, v2f b, v8f c) {
    // V_WMMA_F32_16X16X4_F32 : D = A(16x4) * B(4x16) + C
    return __builtin_amdgcn_wmma_f32_16x16x4_f32(
        false, a, false, b, (short)0, c, false, false);
}

__device__ __forceinline__ float quantv(int q, float x, float y) {
    switch (q) {
        case 0: return x;
        case 1: return y;
        case 2: return x * x;
        case 3: return y * y;
        default: return x * y;
    }
}

__global__ __launch_bounds__(32)
void ssim_tile_kernel(const float* __restrict__ X, const float* __restrict__ Y,
                      float* __restrict__ acc, int level, int H, int S, int tiles)
{
    __shared__ float w[WIN];
    __shared__ float Xp[32][28];
    __shared__ float Yp[32][28];
    __shared__ float WT[16][28];   // WT[n][k] = w[k-n] (banded Toeplitz, zero outside band)
    __shared__ float Tt[16][36];   // intermediate, stored transposed: Tt[col][row]

    const int t = threadIdx.x;

    // Gaussian window (size 11, sigma 1.5), normalized
    if (t < WIN) {
        float d = (float)(t - WIN / 2);
        w[t] = expf(-(d * d) / (2.0f * 1.5f * 1.5f));
    }
    __syncthreads();
    if (t == 0) {
        float s = 0.0f;
        for (int i = 0; i < WIN; ++i) s += w[i];
        float inv = 1.0f / s;
        for (int i = 0; i < WIN; ++i) w[i] *= inv;
    }
    __syncthreads();

    // Build shared weight matrix (branchless band select)
    for (int idx = t; idx < 16 * 28; idx += 32) {
        int n = idx / 28, k = idx - n * 28;
        int d = k - n;
        WT[n][k] = (d >= 0 && d <= 10) ? w[d] : 0.0f;
    }

    const int tilesSq = tiles * tiles;
    const int bc  = blockIdx.x / tilesSq;
    const int rem = blockIdx.x - bc * tilesSq;
    const int ty  = rem / tiles;
    const int tx  = rem - ty * tiles;
    const int oy0 = ty * 16, ox0 = tx * 16;

    const float* Xb = X + (size_t)bc * H * H;
    const float* Yb = Y + (size_t)bc * H * H;

    // Load 32x28 clamped input patches (only rows/cols 0..25 carry weight)
    for (int idx = t; idx < 32 * 28; idx += 32) {
        int r = idx / 28, c = idx - r * 28;
        int gy = oy0 + r; if (gy > H - 1) gy = H - 1;
        int gx = ox0 + c; if (gx > H - 1) gx = H - 1;
        Xp[r][c] = Xb[(size_t)gy * H + gx];
        Yp[r][c] = Yb[(size_t)gy * H + gx];
    }
    __syncthreads();

    const int lanehi = t >> 4;   // 0 or 1
    const int lanelo = t & 15;
    const int fcol   = lanehi * 2;

    // Hoisted loop-invariant fragments:
    // wf[i] serves as B (horizontal: B[k][n]=WT[n][k]) and A (vertical: A[m][j]=WT[m][j])
    v2f wf[7];
    #pragma unroll
    for (int i = 0; i < 7; ++i) {
        wf[i].x = WT[lanelo][4 * i + fcol];
        wf[i].y = WT[lanelo][4 * i + fcol + 1];
    }
    // Raw patch fragments (quantities derived in registers per q)
    v2f xa[2][7], ya[2][7];
    #pragma unroll
    for (int g = 0; g < 2; ++g) {
        #pragma unroll
        for (int i = 0; i < 7; ++i) {
            int r = g * 16 + lanelo, c = 4 * i + fcol;
            xa[g][i].x = Xp[r][c]; xa[g][i].y = Xp[r][c + 1];
            ya[g][i].x = Yp[r][c]; ya[g][i].y = Yp[r][c + 1];
        }
    }

    v8f blur[5];
    #pragma unroll
    for (int q = 0; q < 5; ++q) {
        // ---- horizontal: T(32x16) = P(32x28) * Wh(28x16), two 16-row groups ----
        #pragma unroll
        for (int g = 0; g < 2; ++g) {
            v8f av = {};
            #pragma unroll
            for (int i = 0; i < 7; ++i) {
                v2f a;
                a.x = quantv(q, xa[g][i].x, ya[g][i].x);
                a.y = quantv(q, xa[g][i].y, ya[g][i].y);
                av = wmma_f32_k4(a, wf[i], av);
            }
            // D rows (base..base+7) are contiguous per lane in transposed storage
            int base = g * 16 + lanehi * 8;
            *(float4*)&Tt[lanelo][base]     = make_float4(av[0], av[1], av[2], av[3]);
            *(float4*)&Tt[lanelo][base + 4] = make_float4(av[4], av[5], av[6], av[7]);
        }
        __syncthreads();

        // ---- vertical: out(16x16) = Wv(16x28) * T(28x16) ----
        v8f ov = {};
        #pragma unroll
        for (int i = 0; i < 7; ++i) {
            v2f b;
            b.x = Tt[lanelo][4 * i + fcol];
            b.y = Tt[lanelo][4 * i + fcol + 1];
            ov = wmma_f32_k4(wf[i], b, ov);
        }
        blur[q] = ov;
        __syncthreads();   // Tt reused next quantity
    }

    // ---- SSIM / CS maps + masked tile reduction ----
    float ssim_s = 0.0f, cs_s = 0.0f;
    #pragma unroll
    for (int j = 0; j < 8; ++j) {
        int r = oy0 + j + lanehi * 8;
        int c = ox0 + lanelo;
        float mu1 = blur[0][j], mu2 = blur[1][j];
        float s1  = blur[2][j] - mu1 * mu1;
        float s2  = blur[3][j] - mu2 * mu2;
        float s12 = blur[4][j] - mu1 * mu2;
        float cs   = (2.0f * s12 + SSIM_C2) / (s1 + s2 + SSIM_C2);
        float ssim = ((2.0f * mu1 * mu2 + SSIM_C1) /
                      (mu1 * mu1 + mu2 * mu2 + SSIM_C1)) * cs;
        if (r < S && c < S) { ssim_s += ssim; cs_s += cs; }
    }
    for (int off = 16; off > 0; off >>= 1) {
        ssim_s += __shfl_down(ssim_s, off, 32);
        cs_s   += __shfl_down(cs_s,   off, 32);
    }
    if (t == 0) {
        atomicAdd(&acc[(level * 48 + bc) * 2 + 0], ssim_s);
        atomicAdd(&acc[(level * 48 + bc) * 2 + 1], cs_s);
    }
}

__global__ void avgpool_kernel(const float* __restrict__ gsrc, const float* __restrict__ tsrc,
                               float* __restrict__ gdst, float* __restrict__ tdst,
                               int Hs, int Hd)
{
    int idx = blockIdx.x * blockDim.x + threadIdx.x;
    int n1 = 48 * Hd * Hd;
    if (idx >= 2 * n1) return;
    const float* src = (idx < n1) ? gsrc : tsrc;
    float*       dst = (idx < n1) ? gdst : tdst;
    int i = (idx < n1) ? idx : idx - n1;
    int bc = i / (Hd * Hd);
    int r  = i - bc * Hd * Hd;
    int y = r / Hd, x = r - y * Hd;
    const float* s = src + (size_t)bc * Hs * Hs;
    float v = 0.25f * (s[(size_t)(2 * y) * Hs + 2 * x]     + s[(size_t)(2 * y) * Hs + 2 * x + 1] +
                       s[(size_t)(2 * y + 1) * Hs + 2 * x] + s[(size_t)(2 * y + 1) * Hs + 2 * x + 1]);
    dst[i] = v;
}

__global__ void zero_kernel(float* __restrict__ p, int n) {
    int i = blockIdx.x * blockDim.x + threadIdx.x;
    if (i < n) p[i] = 0.0f;
}

__global__ __launch_bounds__(64)
void finalize_kernel(const float* __restrict__ acc, float* __restrict__ out)
{
    __shared__ float red[64];
    int t = threadIdx.x;
    const float wts[5] = {0.0448f, 0.2856f, 0.3001f, 0.2363f, 0.1333f};
    const int   Ssz[5] = {502, 246, 118, 54, 22};
    float v = 0.0f;
    if (t < 48) {
        float p = 1.0f;
        #pragma unroll
        for (int l = 0; l < 5; ++l) {
            float inv = 1.0f / ((float)Ssz[l] * (float)Ssz[l]);
            float val = (l < 4) ? acc[(l * 48 + t) * 2 + 1]
                                : acc[(l * 48 + t) * 2 + 0];
            val *= inv;
            val = fmaxf(val, 0.0f);
            p *= powf(val, wts[l]);
        }
        v = p;
    }
    red[t] = v;
    __syncthreads();
    for (int s = 32; s > 0; s >>= 1) {
        if (t < s) red[t] += red[t + s];
        __syncthreads();
    }
    if (t == 0) out[0] = 1.0f - red[0] / 48.0f;
}

extern "C" void kernel_launch(void* const* d_in, const int* in_sizes, int n_in,
                              void* d_out, int out_size, void* d_ws, size_t ws_size,
                              hipStream_t stream)
{
    const float* gen = (const float*)d_in[0];
    const float* gt  = (const float*)d_in[1];
    float* out = (float*)d_out;
    float* ws  = (float*)d_ws;

    // workspace layout (floats): [0..479] accumulators, then pyramid buffers
    float* acc = ws;
    size_t off = 512;
    const int Hs[5] = {512, 256, 128, 64, 32};
    const float* G[5]; const float* Tt[5];
    float* bufG[5]; float* bufT[5];
    G[0] = gen; Tt[0] = gt;
    for (int l = 1; l < 5; ++l) {
        size_t n = (size_t)48 * Hs[l] * Hs[l];
        bufG[l] = ws + off; off += n;
        bufT[l] = ws + off; off += n;
        G[l] = bufG[l]; Tt[l] = bufT[l];
    }

    zero_kernel<<<1, 512, 0, stream>>>(acc, 480);

    for (int l = 0; l < 5; ++l) {
        int S = Hs[l] - 10;                // VALID blur output size
        int tiles = (S + 15) / 16;
        int nblk = tiles * tiles * 48;
        ssim_tile_kernel<<<nblk, 32, 0, stream>>>(G[l], Tt[l], acc, l, Hs[l], S, tiles);
        if (l < 4) {
            int Hd = Hs[l + 1];
            int n = 2 * 48 * Hd * Hd;
            avgpool_kernel<<<(n + 255) / 256, 256, 0, stream>>>(
                G[l], Tt[l], bufG[l + 1], bufT[l + 1], Hs[l], Hd);
        }
    }

    finalize_kernel<<<1, 64, 0, stream>>>(acc, out);
}